// GINLayer_3006477107662
// MI455X (gfx1250) — compile-verified
//
#include <hip/hip_runtime.h>
#include <hip/hip_bf16.h>
#include <stdint.h>

typedef float v2f __attribute__((ext_vector_type(2)));
typedef float v8f __attribute__((ext_vector_type(8)));
typedef unsigned int u32x4 __attribute__((ext_vector_type(4)));
typedef int i32x4 __attribute__((ext_vector_type(4)));
typedef int i32x8 __attribute__((ext_vector_type(8)));

#define FEAT   512        // D == H == 512
#define BN_EPS 1e-5f
#define KCHUNK 64         // K rows of W staged per LDS chunk
#define NTILE  64         // N columns per block
#define NCHUNK (FEAT / KCHUNK)

#if defined(__has_builtin)
#if __has_builtin(__builtin_amdgcn_tensor_load_to_lds) && __has_builtin(__builtin_amdgcn_s_wait_tensorcnt)
#define USE_TDM 1
#endif
#endif
#ifndef USE_TDM
#define USE_TDM 0
#endif

// ---------------------------------------------------------------------------
// K0: zero scratch (agg + BN stat accumulators)
// ---------------------------------------------------------------------------
__global__ void zero_f32(float* __restrict__ p, long n) {
    long i = (long)blockIdx.x * blockDim.x + threadIdx.x;
    const long stride = (long)gridDim.x * blockDim.x;
    for (; i < n; i += stride) p[i] = 0.0f;
}

// ---------------------------------------------------------------------------
// K1: edge scatter-add.  agg[dst] += x[src]
// One wave per edge; float4 gathers (coalesced, L2-resident) + f32 atomics.
// ---------------------------------------------------------------------------
__global__ void scatter_add(const float* __restrict__ x,
                            const int* __restrict__ src,
                            const int* __restrict__ dst,
                            float* __restrict__ agg, int E) {
    const int lane = threadIdx.x & 31;
    const int e = blockIdx.x * 8 + (threadIdx.x >> 5);
    if (e >= E) return;
    const int s = src[e];
    const int d = dst[e];
    const float4* xs = (const float4*)(x + (long)s * FEAT);
    float* ag = agg + (long)d * FEAT;
#pragma unroll
    for (int j = 0; j < 4; ++j) {
        const int q = lane + j * 32;        // float4 index, 0..127
        float4 v = xs[q];
        const int f = q * 4;
        atomicAdd(ag + f + 0, v.x);
        atomicAdd(ag + f + 1, v.y);
        atomicAdd(ag + f + 2, v.z);
        atomicAdd(ag + f + 3, v.w);
    }
}

#if USE_TDM
// ---------------------------------------------------------------------------
// Issue one TDM DMA of a KCHUNK x NTILE f32 tile of W (row stride FEAT)
// into LDS at byte offset ldsOff.  D# layout per CDNA5 ISA.
// 6-arg builtin form (clang-23 / therock-10.0 headers).
// ---------------------------------------------------------------------------
__device__ __forceinline__ void tdm_load_tile(unsigned int ldsOff,
                                              unsigned long long gaddr) {
    u32x4 g0;
    g0.x = 1u;                                            // count=1, user desc
    g0.y = ldsOff;                                        // lds_addr [63:32]
    g0.z = (unsigned int)(gaddr & 0xFFFFFFFFull);         // global_addr lo
    g0.w = (unsigned int)((gaddr >> 32) & 0x1FFFFFFull)   // global_addr [56:32]
         | (2u << 30);                                    // type = 2 (image)
    i32x8 g1;
    g1[0] = 2 << 16;                 // data_size=2 (4B); mask=0; no pad/iter
    g1[1] = (FEAT & 0xFFFF) << 16;   // tensor_dim0 lo16  (bits 63:48)
    g1[2] = (FEAT & 0xFFFF) << 16;   // dim0 hi16=0 | tensor_dim1 lo16
    g1[3] = NTILE << 16;             // dim1 hi16=0 | tile_dim0=NTILE
    g1[4] = KCHUNK;                  // tile_dim1=KCHUNK, tile_dim2=0
    g1[5] = FEAT;                    // tensor_dim0_stride lo32
    g1[6] = 0;                       // stride hi16 | tensor_dim1_stride lo16
    g1[7] = 0;
    i32x4 z4 = {0, 0, 0, 0};         // groups 2/3 unused (2D tensor)
    i32x8 z8 = {0, 0, 0, 0, 0, 0, 0, 0};
    __builtin_amdgcn_tensor_load_to_lds(g0, g1, z4, z4, z8, 0);
}
#endif

// ---------------------------------------------------------------------------
// K2/K3: fp32 WMMA GEMM,  out[M,512] = (A (+X)) @ W + bias
// Wave computes a 16(M) x 64(N) strip with 4 accumulators (A-frag reuse x4).
// Block = 8 waves stacked in M -> 128x64 tile. Grid = (ceil(M/128), 512/64).
// B is staged K-chunk by K-chunk into double-buffered LDS via the TDM
// (ds_load fragments thereafter); A streams from global (L2-resident).
// Fragment layouts per CDNA5 ISA (V_WMMA_F32_16X16X4_F32):
//   A (16x4): lane&15 = M row,  kHalf = 2*(lane>=16)
//   B (4x16): lane&15 = N col,  same kHalf split
//   C (16x16): VGPR r holds M = r + 8*(lane>=16), N = lane&15
// ---------------------------------------------------------------------------
template <bool FUSE_ADD_X, bool RELU, bool STATS>
__global__ void gemm_wmma_f32(const float* __restrict__ A,
                              const float* __restrict__ X,
                              const float* __restrict__ W,
                              const float* __restrict__ bias,
                              float* __restrict__ out,
                              float* __restrict__ colsum,
                              float* __restrict__ colsumsq,
                              int M) {
    __shared__ float Bs[2][KCHUNK * NTILE];

    const int lane = threadIdx.x & 31;
    const int wave = threadIdx.x >> 5;              // 0..7
    const int mBase = blockIdx.x * 128 + wave * 16; // wave's M tile origin
    const int nBase = blockIdx.y * NTILE;           // block's N strip origin

    const int mRow  = mBase + (lane & 15);
    const int mClmp = mRow < M ? mRow : (M - 1);    // clamp tail loads
    const int kHalf = (lane >> 4) << 1;             // 0 or 2
    const int nLoc  = lane & 15;

    v8f c0 = {}, c1 = {}, c2 = {}, c3 = {};

    const float* __restrict__ aBase = A + (long)mClmp * FEAT;
    const float* __restrict__ xBase = FUSE_ADD_X ? (X + (long)mClmp * FEAT) : nullptr;

#if USE_TDM
    const unsigned long long wBase =
        (unsigned long long)(uintptr_t)(W + nBase);
    const unsigned int lds0 = (unsigned int)(uintptr_t)(&Bs[0][0]);
    const unsigned int lds1 = (unsigned int)(uintptr_t)(&Bs[1][0]);
    if (wave == 0) tdm_load_tile(lds0, wBase);      // prime chunk 0
#endif

    for (int c = 0; c < NCHUNK; ++c) {
        const float* buf = Bs[c & 1];
#if USE_TDM
        if (wave == 0) {
            if (c + 1 < NCHUNK) {
                tdm_load_tile(((c + 1) & 1) ? lds1 : lds0,
                              wBase + (unsigned long long)(c + 1) *
                                      (KCHUNK * FEAT * sizeof(float)));
                __builtin_amdgcn_s_wait_tensorcnt(1);   // chunk c landed
            } else {
                __builtin_amdgcn_s_wait_tensorcnt(0);
            }
        }
        __syncthreads();                                // chunk c visible to all
#else
        __syncthreads();
        {   // cooperative staging: 256 threads x 4 float4 = 64x64 chunk
            const float* src = W + (long)(c * KCHUNK) * FEAT + nBase;
            float4* dstq = (float4*)Bs[c & 1];
            for (int idx = threadIdx.x; idx < KCHUNK * NTILE / 4; idx += 256) {
                const int r = idx / (NTILE / 4);
                const int q = idx % (NTILE / 4);
                dstq[idx] = *(const float4*)(src + (long)r * FEAT + q * 4);
            }
        }
        __syncthreads();
#endif
        const int k0 = c * KCHUNK;
#pragma unroll 4
        for (int k = 0; k < KCHUNK; k += 4) {
            const int ka = k0 + k + kHalf;
            v2f a;
            a.x = aBase[ka];
            a.y = aBase[ka + 1];
            if (FUSE_ADD_X) {
                a.x += xBase[ka];
                a.y += xBase[ka + 1];
            }
            const float* bp = buf + (k + kHalf) * NTILE + nLoc;
            v2f b0, b1, b2, b3;
            b0.x = bp[0];       b0.y = bp[NTILE];
            b1.x = bp[16];      b1.y = bp[NTILE + 16];
            b2.x = bp[32];      b2.y = bp[NTILE + 32];
            b3.x = bp[48];      b3.y = bp[NTILE + 48];
            c0 = __builtin_amdgcn_wmma_f32_16x16x4_f32(false, a, false, b0, (short)0, c0, false, false);
            c1 = __builtin_amdgcn_wmma_f32_16x16x4_f32(false, a, false, b1, (short)0, c1, false, false);
            c2 = __builtin_amdgcn_wmma_f32_16x16x4_f32(false, a, false, b2, (short)0, c2, false, false);
            c3 = __builtin_amdgcn_wmma_f32_16x16x4_f32(false, a, false, b3, (short)0, c3, false, false);
        }
        __syncthreads();    // all waves done with buf before it is re-DMAed
    }

    // Epilogue: bias (+ReLU) (+BN stats partials) and stores.
    const int mOff = (lane >> 4) << 3;   // 0 or 8
#pragma unroll
    for (int t = 0; t < 4; ++t) {
        v8f c = (t == 0) ? c0 : (t == 1) ? c1 : (t == 2) ? c2 : c3;
        const int col = nBase + t * 16 + nLoc;
        const float bv = bias[col];
        float s = 0.0f, sq = 0.0f;
#pragma unroll
        for (int r = 0; r < 8; ++r) {
            const int row = mBase + mOff + r;
            float v = c[r] + bv;
            if (RELU) v = v > 0.0f ? v : 0.0f;
            if (row < M) {
                out[(long)row * FEAT + col] = v;
                if (STATS) { s += v; sq += v * v; }
            }
        }
        if (STATS) {
            atomicAdd(colsum + col, s);
            atomicAdd(colsumsq + col, sq);
        }
    }
}

// ---------------------------------------------------------------------------
// K4: BatchNorm (batch stats, biased var) + residual, in-place on d_out.
// ---------------------------------------------------------------------------
__global__ void bn_residual(float* __restrict__ h,
                            const float* __restrict__ x,
                            const float* __restrict__ colsum,
                            const float* __restrict__ colsumsq,
                            const float* __restrict__ gamma,
                            const float* __restrict__ beta,
                            int N) {
    const float invN = 1.0f / (float)N;
    const long total = (long)N * FEAT / 4;
    long i = (long)blockIdx.x * blockDim.x + threadIdx.x;
    const long stride = (long)gridDim.x * blockDim.x;
    for (; i < total; i += stride) {
        const long base = i * 4;
        const int col = (int)(base & (FEAT - 1));   // multiple of 4
        float4 hv = ((const float4*)h)[i];
        float4 xv = ((const float4*)x)[i];
        float4 sm = *(const float4*)(colsum + col);
        float4 sq = *(const float4*)(colsumsq + col);
        float4 gm = *(const float4*)(gamma + col);
        float4 bt = *(const float4*)(beta + col);
        float m, var, sc;
        m = sm.x * invN; var = sq.x * invN - m * m; sc = gm.x * __frsqrt_rn(var + BN_EPS);
        hv.x = (hv.x - m) * sc + bt.x + xv.x;
        m = sm.y * invN; var = sq.y * invN - m * m; sc = gm.y * __frsqrt_rn(var + BN_EPS);
        hv.y = (hv.y - m) * sc + bt.y + xv.y;
        m = sm.z * invN; var = sq.z * invN - m * m; sc = gm.z * __frsqrt_rn(var + BN_EPS);
        hv.z = (hv.z - m) * sc + bt.z + xv.z;
        m = sm.w * invN; var = sq.w * invN - m * m; sc = gm.w * __frsqrt_rn(var + BN_EPS);
        hv.w = (hv.w - m) * sc + bt.w + xv.w;
        ((float4*)h)[i] = hv;
    }
}

// ---------------------------------------------------------------------------
extern "C" void kernel_launch(void* const* d_in, const int* in_sizes, int n_in,
                              void* d_out, int out_size, void* d_ws, size_t ws_size,
                              hipStream_t stream) {
    const float* x     = (const float*)d_in[0];
    const int*   eidx  = (const int*)d_in[1];
    const float* W1    = (const float*)d_in[2];
    const float* b1    = (const float*)d_in[3];
    const float* W2    = (const float*)d_in[4];
    const float* b2    = (const float*)d_in[5];
    const float* gamma = (const float*)d_in[6];
    const float* beta  = (const float*)d_in[7];

    const int N = in_sizes[0] / FEAT;   // 10000
    const int E = in_sizes[1] / 2;      // 160000

    // Workspace layout: [agg N*FEAT][colsum FEAT][colsumsq FEAT][h1 N*FEAT]
    float* ws       = (float*)d_ws;
    float* agg      = ws;
    float* colsum   = ws + (long)N * FEAT;
    float* colsumsq = colsum + FEAT;
    float* h1       = colsumsq + FEAT;
    float* out      = (float*)d_out;

    // K0: zero agg + stats (contiguous prefix)
    const long zn = (long)N * FEAT + 2 * FEAT;
    zero_f32<<<2048, 256, 0, stream>>>(ws, zn);

    // K1: scatter-add over edges (src = eidx[0,:], dst = eidx[1,:])
    scatter_add<<<(E + 7) / 8, 256, 0, stream>>>(x, eidx, eidx + E, agg, E);

    // K2: h1 = relu((agg + x) @ W1 + b1)
    dim3 grid((N + 127) / 128, FEAT / NTILE);
    gemm_wmma_f32<true, true, false><<<grid, 256, 0, stream>>>(
        agg, x, W1, b1, h1, nullptr, nullptr, N);

    // K3: out = h1 @ W2 + b2, accumulating per-column sum / sumsq
    gemm_wmma_f32<false, false, true><<<grid, 256, 0, stream>>>(
        h1, nullptr, W2, b2, out, colsum, colsumsq, N);

    // K4: out = BN(out) * gamma + beta + x   (in-place)
    const long quads = (long)N * FEAT / 4;
    bn_residual<<<(int)((quads + 255) / 256), 256, 0, stream>>>(
        out, x, colsum, colsumsq, gamma, beta, N);
}